// Readout_89885075571228
// MI455X (gfx1250) — compile-verified
//
#include <hip/hip_runtime.h>

typedef __attribute__((ext_vector_type(2))) float v2f;
typedef __attribute__((ext_vector_type(8))) float v8f;

#define BATCH     512
#define N_ATOMS   128
#define N_BONDS   256
#define MAX_NBR   8
#define BOND_D    128
#define ATOM_D    128
#define CAT_D     256   // ATOM_D + BOND_D
#define HIDDEN    256
#define CHUNK_M   64                       // atoms per LDS chunk
#define N_CHUNKS  (N_ATOMS / CHUNK_M)      // 2
#define M_TILES   (CHUNK_M / 16)           // 4
#define LDS_STRIDE 260                     // padded row stride (floats): 4*row mod 64 banks, 16B aligned

#ifndef __has_builtin
#define __has_builtin(x) 0
#endif

// Signature established from round-2/3 diagnostics:
//   param0: int __vector(4) addrspace(1)*  (global, 16B vector, non-const)
//   param1: int __vector(4) addrspace(3)*  (LDS, 16B vector)
//   param2/3: immediate int offset / cpol
typedef __attribute__((__vector_size__(16))) int v4i_b128;
typedef __attribute__((address_space(1))) v4i_b128* g128_t;
typedef __attribute__((address_space(3))) v4i_b128* l128_t;

__global__ __launch_bounds__(256)
void readout_wmma_kernel(const float* __restrict__ bond0,      // [B, N_BONDS, BOND_D]
                         const float* __restrict__ atom_feats, // [B, N_ATOMS, ATOM_D]
                         const float* __restrict__ W,          // [CAT_D, HIDDEN] row-major
                         const float* __restrict__ bias,       // [HIDDEN]
                         const int*   __restrict__ nbrs,       // [B, N_ATOMS, MAX_NBR]
                         float*       __restrict__ out)        // [B, HIDDEN]
{
    __shared__ float cat[CHUNK_M * LDS_STRIDE];   // [64][260] staged cat chunk (~66.5 KB)
    __shared__ float lds_out[HIDDEN];             // per-batch output accumulator
    __shared__ float lds_bias[HIDDEN];

    const int b    = blockIdx.x;
    const int tid  = threadIdx.x;
    const int lane = tid & 31;
    const int wave = tid >> 5;                    // 0..7

    const float* bondB = bond0      + (size_t)b * N_BONDS * BOND_D;
    const float* atomB = atom_feats + (size_t)b * N_ATOMS * ATOM_D;
    const int*   nbrB  = nbrs       + (size_t)b * N_ATOMS * MAX_NBR;

    if (tid < HIDDEN) { lds_out[tid] = 0.0f; lds_bias[tid] = bias[tid]; }
    __syncthreads();

    for (int chunk = 0; chunk < N_CHUNKS; ++chunk) {
        const int a0 = chunk * CHUNK_M;

        // ---------- Phase 1: build cat chunk [64][256] in LDS ----------
        // 4 threads per atom; each thread owns 32 contiguous columns (8 x float4).
        {
            const int aloc = tid >> 2;                 // 0..63
            const int cg   = (tid & 3) * 32;           // column-group base within 128
            const int a    = a0 + aloc;

            // --- atom features -> cat cols [0,128): async direct-to-LDS copy ---
#if __has_builtin(__builtin_amdgcn_global_load_async_to_lds_b128)
            {
                g128_t g = (g128_t)(atomB + (size_t)a * ATOM_D + cg);
                l128_t l = (l128_t)&cat[aloc * LDS_STRIDE + cg];
                // INST_OFFSET applies to both global and LDS addresses (ISA 08 §4.4)
                __builtin_amdgcn_global_load_async_to_lds_b128(g, l,   0, 0);
                __builtin_amdgcn_global_load_async_to_lds_b128(g, l,  16, 0);
                __builtin_amdgcn_global_load_async_to_lds_b128(g, l,  32, 0);
                __builtin_amdgcn_global_load_async_to_lds_b128(g, l,  48, 0);
                __builtin_amdgcn_global_load_async_to_lds_b128(g, l,  64, 0);
                __builtin_amdgcn_global_load_async_to_lds_b128(g, l,  80, 0);
                __builtin_amdgcn_global_load_async_to_lds_b128(g, l,  96, 0);
                __builtin_amdgcn_global_load_async_to_lds_b128(g, l, 112, 0);
            }
#else
            {
                const float4* asrc = (const float4*)(atomB + (size_t)a * ATOM_D + cg);
                float4* adst = (float4*)(&cat[aloc * LDS_STRIDE + cg]);
                #pragma unroll
                for (int j = 0; j < 8; ++j) adst[j] = asrc[j];
            }
#endif

            // --- gathered bond message sum -> cat cols [128,256) (overlaps async) ---
            float4 acc[8];
            #pragma unroll
            for (int j = 0; j < 8; ++j) acc[j] = make_float4(0.f, 0.f, 0.f, 0.f);
            #pragma unroll
            for (int nb = 0; nb < MAX_NBR; ++nb) {
                const int bd = nbrB[a * MAX_NBR + nb];
                const float4* bsrc = (const float4*)(bondB + (size_t)bd * BOND_D + cg);
                #pragma unroll
                for (int j = 0; j < 8; ++j) {
                    float4 v = bsrc[j];
                    acc[j].x += v.x; acc[j].y += v.y; acc[j].z += v.z; acc[j].w += v.w;
                }
            }
            float4* mdst = (float4*)(&cat[aloc * LDS_STRIDE + ATOM_D + cg]);
            #pragma unroll
            for (int j = 0; j < 8; ++j) mdst[j] = acc[j];

#if __has_builtin(__builtin_amdgcn_global_load_async_to_lds_b128)
#if __has_builtin(__builtin_amdgcn_s_wait_asynccnt)
            __builtin_amdgcn_s_wait_asynccnt(0);
#else
            asm volatile("s_wait_asynccnt 0" ::: "memory");
#endif
#endif
        }
        __syncthreads();

        // ---------- Phase 2: WMMA GEMM (cat[64x256] @ W[256x256]) ----------
        // Wave w owns N-tiles {w, w+8}; all 4 M-tiles of the chunk share B frags,
        // both N-tiles share A frags: 8 WMMA per 4 global b32 + 4 ds b64 per K-step.
        {
            const int l15   = lane & 15;
            const int khalf = (lane >> 4) * 2;         // lanes 0-15 -> k,k+1 ; 16-31 -> k+2,k+3
            const int n0    = wave * 16 + l15;
            const int n1    = (wave + 8) * 16 + l15;

            const float* catRow0 = &cat[(0 * 16 + l15) * LDS_STRIDE];
            const float* catRow1 = &cat[(1 * 16 + l15) * LDS_STRIDE];
            const float* catRow2 = &cat[(2 * 16 + l15) * LDS_STRIDE];
            const float* catRow3 = &cat[(3 * 16 + l15) * LDS_STRIDE];

            v8f acc00 = {}, acc01 = {}, acc02 = {}, acc03 = {};
            v8f acc10 = {}, acc11 = {}, acc12 = {}, acc13 = {};

            #pragma unroll 2
            for (int k = 0; k < CAT_D; k += 4) {
                const int kk = k + khalf;
                // A fragments (16x4 f32): lane holds cat[row][kk], cat[row][kk+1]
                v2f a0 = *(const v2f*)(catRow0 + kk);
                v2f a1 = *(const v2f*)(catRow1 + kk);
                v2f a2 = *(const v2f*)(catRow2 + kk);
                v2f a3 = *(const v2f*)(catRow3 + kk);
                // B fragments (4x16 f32): lane holds W[kk][n], W[kk+1][n]
                v2f b0, b1;
                b0.x = W[(size_t)(kk    ) * HIDDEN + n0];
                b0.y = W[(size_t)(kk + 1) * HIDDEN + n0];
                b1.x = W[(size_t)(kk    ) * HIDDEN + n1];
                b1.y = W[(size_t)(kk + 1) * HIDDEN + n1];

                acc00 = __builtin_amdgcn_wmma_f32_16x16x4_f32(false, a0, false, b0, (short)0, acc00, false, false);
                acc01 = __builtin_amdgcn_wmma_f32_16x16x4_f32(false, a1, false, b0, (short)0, acc01, false, false);
                acc02 = __builtin_amdgcn_wmma_f32_16x16x4_f32(false, a2, false, b0, (short)0, acc02, false, false);
                acc03 = __builtin_amdgcn_wmma_f32_16x16x4_f32(false, a3, false, b0, (short)0, acc03, false, false);
                acc10 = __builtin_amdgcn_wmma_f32_16x16x4_f32(false, a0, false, b1, (short)0, acc10, false, false);
                acc11 = __builtin_amdgcn_wmma_f32_16x16x4_f32(false, a1, false, b1, (short)0, acc11, false, false);
                acc12 = __builtin_amdgcn_wmma_f32_16x16x4_f32(false, a2, false, b1, (short)0, acc12, false, false);
                acc13 = __builtin_amdgcn_wmma_f32_16x16x4_f32(false, a3, false, b1, (short)0, acc13, false, false);
            }

            // bias + relu + reduce over the 64 M rows this lane-column touches,
            // then accumulate into the per-batch output accumulator.
            // C/D layout: VGPR r = row r (lanes 0-15) / row r+8 (lanes 16-31), col fixed per lane.
            const float bias0 = lds_bias[n0];
            const float bias1 = lds_bias[n1];
            float s0 = 0.f, s1 = 0.f;
            #pragma unroll
            for (int i = 0; i < 8; ++i) {
                s0 += fmaxf(acc00[i] + bias0, 0.f) + fmaxf(acc01[i] + bias0, 0.f)
                    + fmaxf(acc02[i] + bias0, 0.f) + fmaxf(acc03[i] + bias0, 0.f);
                s1 += fmaxf(acc10[i] + bias1, 0.f) + fmaxf(acc11[i] + bias1, 0.f)
                    + fmaxf(acc12[i] + bias1, 0.f) + fmaxf(acc13[i] + bias1, 0.f);
            }
            atomicAdd(&lds_out[n0], s0);   // ds_add_f32
            atomicAdd(&lds_out[n1], s1);
        }
        __syncthreads();   // protect cat before next chunk overwrites it
    }

    if (tid < HIDDEN) out[(size_t)b * HIDDEN + tid] = lds_out[tid];
}

extern "C" void kernel_launch(void* const* d_in, const int* in_sizes, int n_in,
                              void* d_out, int out_size, void* d_ws, size_t ws_size,
                              hipStream_t stream) {
    // setup_inputs order: bond_repr, atom_feats, W_o, b_o, nbrs
    const float* bond  = (const float*)d_in[0];   // [1, B, N_BONDS, BOND_D] -> leading 1 is flat
    const float* atomf = (const float*)d_in[1];   // [B, N_ATOMS, ATOM_D]
    const float* W     = (const float*)d_in[2];   // [CAT_D, HIDDEN]
    const float* bo    = (const float*)d_in[3];   // [HIDDEN]
    const int*   nbr   = (const int*)d_in[4];     // [B, N_ATOMS, MAX_NBR]
    float*       outp  = (float*)d_out;           // [B, HIDDEN]

    readout_wmma_kernel<<<dim3(BATCH), dim3(256), 0, stream>>>(bond, atomf, W, bo, nbr, outp);
}